// GraphConv_15487652069473
// MI455X (gfx1250) — compile-verified
//
#include <hip/hip_runtime.h>

#define N_NODES 50000
#define N_EDGES 350000
#define IN_C    256
#define OUT_C   256
#define NT      7
#define K_DIM   (NT * IN_C)     // 1792
#define K_CHUNK 64
#define NCHUNK  (K_DIM / K_CHUNK)   // 28
#define M_TILE  32
#define LDS_PITCH 68            // 64 + 4 pad -> conflict-free ds_load_b64

typedef __attribute__((ext_vector_type(2))) float v2f;
typedef __attribute__((ext_vector_type(8))) float v8f;

// ---------------------------------------------------------------------------
// 1) zero the workspace (sums + counts) -- float4 stores, one pass
// ---------------------------------------------------------------------------
__global__ void zero_ws_kernel(float4* __restrict__ ws, long total4) {
    long i = (long)blockIdx.x * blockDim.x + threadIdx.x;
    if (i < total4) {
        ws[i] = make_float4(0.f, 0.f, 0.f, 0.f);
    }
}

// ---------------------------------------------------------------------------
// 2) edge scatter: one wave32 per edge.
//    Each lane reads 2x float4 of x[col] (256 floats / wave, fully coalesced)
//    and issues 8 hardware global_atomic_add_f32 into the (row, type) segment.
// ---------------------------------------------------------------------------
__global__ __launch_bounds__(256)
void scatter_kernel(const float* __restrict__ x,
                    const int*   __restrict__ edge_index,  // [2, E]
                    const int*   __restrict__ edge_type,   // [E]
                    float* __restrict__ sums,              // [N*7, 256]
                    float* __restrict__ cnt) {             // [N*7]
    int e    = blockIdx.x * (blockDim.x >> 5) + (threadIdx.x >> 5);
    int lane = threadIdx.x & 31;
    if (e >= N_EDGES) return;

    int  row = edge_index[e];
    int  col = edge_index[N_EDGES + e];
    int  t   = edge_type[e];
    long seg = (long)row * NT + t;

    const float4* xs = (const float4*)(x + (long)col * IN_C);
    float4 v0 = xs[lane];        // channels lane*4 .. lane*4+3
    float4 v1 = xs[lane + 32];   // channels 128 + lane*4 ..

    float* dst = sums + seg * IN_C;
    int c = lane * 4;
    unsafeAtomicAdd(dst + c + 0,       v0.x);
    unsafeAtomicAdd(dst + c + 1,       v0.y);
    unsafeAtomicAdd(dst + c + 2,       v0.z);
    unsafeAtomicAdd(dst + c + 3,       v0.w);
    unsafeAtomicAdd(dst + 128 + c + 0, v1.x);
    unsafeAtomicAdd(dst + 128 + c + 1, v1.y);
    unsafeAtomicAdd(dst + 128 + c + 2, v1.z);
    unsafeAtomicAdd(dst + 128 + c + 3, v1.w);

    if (lane == 0) unsafeAtomicAdd(cnt + seg, 1.0f);
}

// ---------------------------------------------------------------------------
// 3) counts -> reciprocals in place: inv = 1 / max(cnt, 1)
// ---------------------------------------------------------------------------
__global__ void inv_kernel(float* __restrict__ cnt, int n) {
    int i = blockIdx.x * blockDim.x + threadIdx.x;
    if (i < n) cnt[i] = 1.0f / fmaxf(cnt[i], 1.0f);
}

// ---------------------------------------------------------------------------
// 4) WMMA f32 GEMM: out[50000,256] = (sums * inv) [50000,1792] @ W [1792,256]
//    Block = 512 thr (16 waves) owns 32 rows (two 16x16 M-tiles per wave);
//    wave w owns columns w*16..w*16+15 -> full N=256 per block.
//    K chunked by 64 (never crosses an edge-type slab since 256 % 64 == 0).
//    Double-buffered LDS pipeline: chunk k+1's HBM loads are issued before
//    chunk k's 32 WMMAs, stored to the alternate buffer after, 1 barrier/chunk.
//    Two accumulators share each B fragment -> half the B traffic per FLOP.
// ---------------------------------------------------------------------------
__global__ __launch_bounds__(512)
void gemm_kernel(const float* __restrict__ sums,   // [N*7, 256]
                 const float* __restrict__ inv,    // [N*7]
                 const float* __restrict__ w,      // [1792, 256]
                 float* __restrict__ out) {        // [N, 256]
    __shared__ float A_lds[2][M_TILE * LDS_PITCH];

    const int tid     = threadIdx.x;
    const int waveId  = tid >> 5;
    const int lane    = tid & 31;
    const int rowBase = blockIdx.x * M_TILE;
    const int colBase = waveId * 16;
    const int m       = lane & 15;          // fragment row / output col
    const int kOff    = (lane >> 4) << 1;   // 0 for lanes 0-15, 2 for 16-31

    // staging role: thread -> (row sr, 4 channels at sc); clamp row so the
    // tail block still issues valid addresses (WMMA needs full EXEC anyway).
    const int sr   = tid >> 4;              // 0..31
    const int sc   = (tid & 15) * 4;        // 0,4,..,60
    const int srow = min(rowBase + sr, N_NODES - 1);

    v8f acc0 = {};
    v8f acc1 = {};

    // ---- prologue: stage chunk 0 into buffer 0 ----
    {
        const long seg = (long)srow * NT;                   // t = 0, cbase = 0
        const float4 v = *(const float4*)(sums + seg * IN_C + sc);
        const float  s = inv[seg];
        *(float4*)(&A_lds[0][sr * LDS_PITCH + sc]) =
            make_float4(v.x * s, v.y * s, v.z * s, v.w * s);
    }
    __syncthreads();

    for (int chunk = 0; chunk < NCHUNK; ++chunk) {
        const int buf = chunk & 1;

        // ---- issue next chunk's HBM loads early (overlap with WMMAs) ----
        float4 nv = make_float4(0.f, 0.f, 0.f, 0.f);
        float  ns = 0.f;
        if (chunk + 1 < NCHUNK) {
            const int  nk   = (chunk + 1) * K_CHUNK;
            const long nseg = (long)srow * NT + (nk >> 8);
            nv = *(const float4*)(sums + nseg * IN_C + (nk & 255) + sc);
            ns = inv[nseg];
        }

        // ---- 16 k-steps x 2 M-tiles = 32 WMMAs over this chunk ----
        const int kbase = chunk * K_CHUNK;
        const float* wp = w + (long)kbase * OUT_C + colBase + m;
        const float* a0 = &A_lds[buf][m * LDS_PITCH + kOff];
        const float* a1 = &A_lds[buf][(m + 16) * LDS_PITCH + kOff];
#pragma unroll
        for (int kk = 0; kk < K_CHUNK; kk += 4) {
            v2f aT0, aT1, b;
            aT0.x = a0[kk];                 // ds_load_b64, conflict-free
            aT0.y = a0[kk + 1];
            aT1.x = a1[kk];
            aT1.y = a1[kk + 1];
            const float* br = wp + (long)(kk + kOff) * OUT_C;
            b.x = br[0];                    // W[k][n]   (L2-resident)
            b.y = br[OUT_C];                // W[k+1][n]
            acc0 = __builtin_amdgcn_wmma_f32_16x16x4_f32(
                false, aT0, false, b, (short)0, acc0, false, false);
            acc1 = __builtin_amdgcn_wmma_f32_16x16x4_f32(
                false, aT1, false, b, (short)0, acc1, false, false);
        }

        // ---- commit next chunk to the alternate buffer ----
        if (chunk + 1 < NCHUNK) {
            *(float4*)(&A_lds[buf ^ 1][sr * LDS_PITCH + sc]) =
                make_float4(nv.x * ns, nv.y * ns, nv.z * ns, nv.w * ns);
        }
        __syncthreads();
    }

    // ---- store: VGPR r -> row (r + 8*(lane>=16)), col = lane&15 ----
    const int mrow = (lane >> 4) << 3;
    float* op0 = out + (long)(rowBase + mrow) * OUT_C + colBase + m;
#pragma unroll
    for (int r = 0; r < 8; ++r) {
        op0[(long)r * OUT_C] = acc0[r];
    }
    if (rowBase + 16 < N_NODES) {           // uniform guard for tail block
        float* op1 = out + (long)(rowBase + 16 + mrow) * OUT_C + colBase + m;
#pragma unroll
        for (int r = 0; r < 8; ++r) {
            op1[(long)r * OUT_C] = acc1[r];
        }
    }
}

// ---------------------------------------------------------------------------
extern "C" void kernel_launch(void* const* d_in, const int* in_sizes, int n_in,
                              void* d_out, int out_size, void* d_ws, size_t ws_size,
                              hipStream_t stream) {
    const float* x          = (const float*)d_in[0];
    const int*   edge_index = (const int*)  d_in[1];
    const int*   edge_type  = (const int*)  d_in[2];
    const float* weights    = (const float*)d_in[3];
    float*       out        = (float*)d_out;

    float* sums = (float*)d_ws;                                  // N*7*256 floats
    float* cnt  = sums + (size_t)N_NODES * NT * IN_C;            // N*7 floats

    // 1) zero sums + counts
    const long totalFloats = (long)N_NODES * NT * IN_C + (long)N_NODES * NT;
    const long total4      = totalFloats / 4;                    // divisible
    zero_ws_kernel<<<(int)((total4 + 255) / 256), 256, 0, stream>>>(
        (float4*)d_ws, total4);

    // 2) scatter-add (8 wave32 edges per 256-thread block)
    scatter_kernel<<<(N_EDGES + 7) / 8, 256, 0, stream>>>(
        x, edge_index, edge_type, sums, cnt);

    // 3) reciprocal counts
    inv_kernel<<<(N_NODES * NT + 255) / 256, 256, 0, stream>>>(cnt, N_NODES * NT);

    // 4) fused mean + projection via V_WMMA_F32_16X16X4_F32
    gemm_kernel<<<(N_NODES + M_TILE - 1) / M_TILE, 512, 0, stream>>>(
        sums, cnt, weights, out);
}